// HybridCoordinateBlockStepper_81071802680054
// MI455X (gfx1250) — compile-verified
//
#include <hip/hip_runtime.h>

// ---------------------------------------------------------------------------
// HybridCoordinateBlockStepper for MI455X (gfx1250, wave32).
//
// Memory-bound fused primitive-equation tendency kernel:
//   unique HBM traffic ~0.8 GB -> ~35us floor @ 23.3 TB/s; AI ~5 FLOP/B.
// Single pass, no HBM intermediates. Vertical C_half cumsum is done as a
// triangular f32 matmul on the wave's 32 columns with V_WMMA_F32_16X16X4_F32
// (exact f32, layout staged through per-wave LDS).
// A_IF == 0 is exploited: p = b_mid*ps, dp = -0.04*ps, log-pressure ratios
// are per-level constants (computed once per block into LDS).
// ---------------------------------------------------------------------------

#define NLAT 721
#define NLON 1440
#define KLEV 24
#define HW   (NLAT * NLON)          // 1,038,240 = 224 * 4635 exactly
#define BLOCK_T 224                 // 7 waves of 32
#define NWAVE   7

#define R_EARTH 6371000.0f
#define R_GAS   287.0f
#define C_P     1004.0f
#define OMEGA_E 7.292e-05f
#define PI_F    3.14159265358979323846f
#define DTH     (PI_F / 720.0f)     // == DTHETA == DLAM

typedef float v2f __attribute__((ext_vector_type(2)));
typedef float v8f __attribute__((ext_vector_type(8)));

__device__ __forceinline__ float bmid(int k) { return 0.98f - 0.04f * (float)k; }

// Inclusive cumsum over k (24 levels) of S for 32 columns of one wave:
//   OUT[m][n] = sum_{k<=m} S[k][n]   via  D = L(24x24, lower-tri ones) * S.
// Uses V_WMMA_F32_16X16X4_F32: 2 M-tiles x 2 N-tiles x 6 K-steps.
// St : per-wave LDS [32 cols][24 k]  (column-major per lane, so B's (k,k+1)
//      pair is contiguous).   CS : per-wave LDS [24 rows][32 cols].
__device__ __forceinline__ void wave_cumsum24(const float* St, float* CS, int lane) {
  const int half = lane >> 4;       // A/B: lanes 16-31 hold K-rows offset by 2
  const int nlo  = lane & 15;
#pragma unroll
  for (int mb = 0; mb < 2; ++mb) {
    const int mrow = mb * 16 + nlo;             // A-matrix row held by this lane
#pragma unroll
    for (int nb = 0; nb < 2; ++nb) {
      const int n = nb * 16 + nlo;              // B/D column held by this lane
      v8f c = {0.f, 0.f, 0.f, 0.f, 0.f, 0.f, 0.f, 0.f};
#pragma unroll
      for (int ks = 0; ks < 6; ++ks) {
        const int ka = ks * 4 + 2 * half;
        v2f a, b;
        a.x = (ka     <= mrow) ? 1.0f : 0.0f;   // L[m][k] = (k <= m)
        a.y = (ka + 1 <= mrow) ? 1.0f : 0.0f;
        b.x = St[n * 24 + ka];
        b.y = St[n * 24 + ka + 1];
        // D = A x B + C, full f32.  (neg_a, A, neg_b, B, c_mod, C, reuse_a, reuse_b)
        c = __builtin_amdgcn_wmma_f32_16x16x4_f32(false, a, false, b, (short)0, c,
                                                  false, false);
      }
      // D layout: VGPR r -> row (mb*16 + 8*half + r), col n.
#pragma unroll
      for (int r = 0; r < 8; ++r) {
        const int row = mb * 16 + half * 8 + r;
        if (row < KLEV) CS[row * 32 + n] = c[r];
      }
    }
  }
}

__global__ __launch_bounds__(BLOCK_T)
void hybrid_step_kernel(const float* __restrict__ uv, const float* __restrict__ T,
                        const float* __restrict__ q,  const float* __restrict__ ps,
                        float* __restrict__ out) {
  __shared__ float s_lrc[KLEV];                 // R-less log pressure ratios
  __shared__ float s_S [NWAVE][32 * 24];        // [col][k] staging for WMMA B
  __shared__ float s_CS[NWAVE][24 * 32];        // [k][col] cumsum result

  const int tix  = threadIdx.x;
  const int lane = tix & 31;
  const int wv   = tix >> 5;
  const int t    = blockIdx.x * BLOCK_T + tix;  // grid covers H*W exactly
  const int h    = t / NLON;
  const int w    = t - h * NLON;

  // per-level constants: lrc[k] = ln(b_mid[k]/b_mid[k+1])  (A_IF==0)
  if (tix < KLEV) {
    s_lrc[tix] = (tix < KLEV - 1) ? logf(bmid(tix) / bmid(tix + 1)) : 0.0f;
  }
  __syncthreads();

  // ---- geometry (clamped indices reproduce one-sided boundary stencils) ----
  const int wm = (w == 0)        ? NLON - 1 : w - 1;
  const int wp = (w == NLON - 1) ? 0        : w + 1;
  const int hm = (h > 0)        ? h - 1 : 0;
  const int hp = (h < NLAT - 1) ? h + 1 : NLAT - 1;

  const float colat = PI_F * (float)h / 720.0f;
  const float c0    = fmaxf(sinf(colat), 1e-3f);                 // cos(lat)
  const float cmv   = fmaxf(sinf(PI_F * (float)hm / 720.0f), 1e-3f);
  const float cpv   = fmaxf(sinf(PI_F * (float)hp / 720.0f), 1e-3f);
  const float fcor  = 2.0f * OMEGA_E * cosf(colat);

  const float dphc  = -1.0f / ((float)(hp - hm) * DTH);          // _dphi factor
  const float ilamc = (1.0f / (2.0f * DTH)) / (R_EARTH * c0);    // _dlam/(R c)
  const float iphc  = dphc / R_EARTH;                            // _dphi/R
  const float iphdv = dphc / (R_EARTH * c0);                     // _dphi/(R c)

  const int hw0 = h  * NLON + w;
  const int oWm = h  * NLON + wm, oWp = h  * NLON + wp;
  const int oHm = hm * NLON + w,  oHp = hp * NLON + w;

  const float ps0 = ps[hw0];
  const float invps0 = 1.0f / ps0;
  const float gpsx = (ps[oWp] - ps[oWm]) * ilamc;
  const float gpsy = (ps[oHp] - ps[oHm]) * iphc;

  const float2* uv2 = (const float2*)uv;

  // ---- pass 1: own column + divergence; stage s_k for the WMMA cumsum ----
  float U[KLEV], V[KLEV], Tt[KLEV], Qt[KLEV];
  float ssum = 0.0f;
  float* Sme = &s_S[wv][lane * 24];
#pragma unroll
  for (int k = 0; k < KLEV; ++k) {
    const int base = k * HW;
    const float2 c  = uv2[base + hw0];
    const float2 xm = uv2[base + oWm];
    const float2 xp = uv2[base + oWp];
    const float2 ym = uv2[base + oHm];
    const float2 yp = uv2[base + oHp];
    U[k] = c.x; V[k] = c.y;
    Tt[k] = T[base + hw0];
    Qt[k] = q[base + hw0];
    const float divg = (xp.x - xm.x) * ilamc + (yp.y * cpv - ym.y * cmv) * iphdv;
    const float vg   = c.x * gpsx + c.y * gpsy;
    const float s    = -0.04f * (ps0 * divg + vg);   // dp*div + db*vgps, db=-0.04
    Sme[k] = s;
    ssum  += s;
  }
  const float dpsdt = -ssum;
  __syncthreads();

  // ---- cumsum over k via V_WMMA_F32_16X16X4_F32 (24 WMMAs / wave) ----
  wave_cumsum24(&s_S[wv][0], &s_CS[wv][0], lane);
  __syncthreads();

  // ---- pass 2: tendencies. Running phi recurrences for own + 4 halo cols ----
  float phi0 = 0.f, phiWm = 0.f, phiWp = 0.f, phiHm = 0.f, phiHp = 0.f;
  float TpWm = 0.f, TpWp = 0.f, TpHm = 0.f, TpHp = 0.f;
  float chPrev = 0.0f;                                  // C_half[0] = 0
  const float rcp = R_GAS / C_P;
  const float* CSme = &s_CS[wv][0];
  float2* out_duv = (float2*)out;

#pragma unroll
  for (int k = 0; k < KLEV; ++k) {
    if (k > 0) {                                        // phi_k = phi_{k-1}+R*T*lrc
      const float rl = R_GAS * s_lrc[k - 1];
      phi0  += rl * Tt[k - 1];
      phiWm += rl * TpWm;  phiWp += rl * TpWp;
      phiHm += rl * TpHm;  phiHp += rl * TpHp;
    }
    const int base = k * HW;
    const float2 xm = uv2[base + oWm];
    const float2 xp = uv2[base + oWp];
    const float2 ym = uv2[base + oHm];
    const float2 yp = uv2[base + oHp];
    const float TWm = T[base + oWm], TWp = T[base + oWp];
    const float THm = T[base + oHm], THp = T[base + oHp];
    const float qWm = q[base + oWm], qWp = q[base + oWp];
    const float qHm = q[base + oHm], qHp = q[base + oHp];

    const float u0 = U[k], v0 = V[k], T0 = Tt[k], q0 = Qt[k];
    const float zeta  = (xp.y - xm.y) * ilamc - (yp.x * cpv - ym.x * cmv) * iphdv;
    const float avort = zeta + fcor;

    // grad(KE + phi): only halo values enter the centered stencils
    const float kpWm = 0.5f * (xm.x * xm.x + xm.y * xm.y) + phiWm;
    const float kpWp = 0.5f * (xp.x * xp.x + xp.y * xp.y) + phiWp;
    const float kpHm = 0.5f * (ym.x * ym.x + ym.y * ym.y) + phiHm;
    const float kpHp = 0.5f * (yp.x * yp.x + yp.y * yp.y) + phiHp;
    const float gkx = (kpWp - kpWm) * ilamc;
    const float gky = (kpHp - kpHm) * iphc;

    // C_half[k+1] = -(cumS[k] + (b_if[k+1]-1)*dpsdt);  Cmid = mean of halves
    const float cumS   = CSme[k * 32 + lane];
    const float chNext = -(cumS + (-0.04f * (float)(k + 1)) * dpsdt);
    const float Cmid   = 0.5f * (chPrev + chNext);
    chPrev = chNext;

    const float bmk = bmid(k);
    const float vg  = u0 * gpsx + v0 * gpsy;
    const float omega = bmk * (dpsdt + vg) + Cmid;

    const int kpp = (k < KLEV - 1) ? k + 1 : k;         // compile-time per iter
    const int kmm = (k > 0)        ? k - 1 : 0;
    const float cdp = invps0 * (1.0f / (-0.04f * (float)(kpp - kmm)));
    const float ddpU = (U [kpp] - U [kmm]) * cdp;
    const float ddpV = (V [kpp] - V [kmm]) * cdp;
    const float ddpT = (Tt[kpp] - Tt[kmm]) * cdp;
    const float ddpQ = (Qt[kpp] - Qt[kmm]) * cdp;

    const float pgf = R_GAS * T0 * invps0;              // R*T*bm/p = R*T/ps

    const float du = avort * v0 - gkx - pgf * gpsx - Cmid * ddpU;
    const float dv = -avort * u0 - gky - pgf * gpsy - Cmid * ddpV;

    const float gTx = (TWp - TWm) * ilamc, gTy = (THp - THm) * iphc;
    const float gqx = (qWp - qWm) * ilamc, gqy = (qHp - qHm) * iphc;
    const float dT = -(u0 * gTx + v0 * gTy) - Cmid * ddpT
                   + rcp * T0 * omega * (invps0 * (1.0f / bmk));
    const float dq = -(u0 * gqx + v0 * gqy) - Cmid * ddpQ;

    out_duv[base + hw0] = make_float2(du, dv);
    out[2 * KLEV * HW + base + hw0] = dT;
    out[3 * KLEV * HW + base + hw0] = dq;

    TpWm = TWm; TpWp = TWp; TpHm = THm; TpHp = THp;
  }
  out[4 * KLEV * HW + hw0] = dpsdt;
}

extern "C" void kernel_launch(void* const* d_in, const int* in_sizes, int n_in,
                              void* d_out, int out_size, void* d_ws, size_t ws_size,
                              hipStream_t stream) {
  (void)in_sizes; (void)n_in; (void)d_ws; (void)ws_size; (void)out_size;
  const float* uv = (const float*)d_in[0];
  const float* T  = (const float*)d_in[1];
  const float* q  = (const float*)d_in[2];
  const float* ps = (const float*)d_in[3];
  float* out = (float*)d_out;
  // 721*1440 = 1,038,240 = 224 * 4635 : exact cover, no tail divergence
  hipLaunchKernelGGL(hybrid_step_kernel, dim3(HW / BLOCK_T), dim3(BLOCK_T), 0, stream,
                     uv, T, q, ps, out);
}